// DiffeqSolver_5746666242687
// MI455X (gfx1250) — compile-verified
//
#include <hip/hip_runtime.h>

// CDNA5 / gfx1250: wave32, WMMA 16x16. V_WMMA_F32_16X16X4_F32 (fp32 in/acc)
// for numerical fidelity of the 127-step Adams-Bashforth recurrence.
// Weights LDS-resident; hidden tiles processed in pairs with software-
// pipelined LDS B-fragment loads to hide ds latency at 1 wave/SIMD.

typedef float v2f __attribute__((ext_vector_type(2)));
typedef float v8f __attribute__((ext_vector_type(8)));

#define WAVES 4
#define BLOCK (WAVES * 32)

// ---- LDS layout (in floats) ----
#define W1I_OFF 0            // 32 pairs x 512 cols x 2 floats = 32768 (128KB)
#define W2I_OFF 32768        // 256 pairs x 64 cols x 2 floats = 32768 (128KB)
#define B1_OFF  65536        // 512
#define B2_OFF  66048        // 64
#define PW_OFF  66112        // per-wave staging
#define YST     0            // 16x64 row-major y stage      (1024 floats)
#define HST     1024         // 2x 16x16 hidden tile stage   (512 floats)
#define DTB     1536         // dt table (T-1)x16, cap 2048  (2048 floats)
#define PW_SIZE 3584
#define SMEM_FLOATS (PW_OFF + WAVES * PW_SIZE)   // 80448 floats = 321792 B <= 320KB

__device__ __forceinline__ v8f wmma4(v2f a, v2f b, v8f c) {
  // D = A(16x4,f32) * B(4x16,f32) + C(16x16,f32) -> v_wmma_f32_16x16x4_f32
  return __builtin_amdgcn_wmma_f32_16x16x4_f32(false, a, false, b, (short)0, c,
                                               false, false);
}

__device__ __forceinline__ v8f splat8(float x) {
  v8f v = {x, x, x, x, x, x, x, x};
  return v;
}

__device__ __forceinline__ float fast_tanh(float x) {
  // tanh(x) = (e-1)/(e+1), e = exp(2x) via v_exp_f32
  float xc = fminf(fmaxf(x, -10.0f), 10.0f);
  float e  = __builtin_amdgcn_exp2f(xc * 2.8853900817779268f);
  return (e - 1.0f) * __builtin_amdgcn_rcpf(e + 1.0f);
}

__global__ __launch_bounds__(BLOCK, 1)
void node_ab4_kernel(const float* __restrict__ ts, const float* __restrict__ hxs,
                     const float* __restrict__ masks, const float* __restrict__ W1,
                     const float* __restrict__ b1, const float* __restrict__ W2,
                     const float* __restrict__ b2, float* __restrict__ out,
                     int N, int T) {
  extern __shared__ float smem[];
  float* w1i = smem + W1I_OFF;
  float* w2i = smem + W2I_OFF;
  float* b1s = smem + B1_OFF;
  float* b2s = smem + B2_OFF;

  const int tid  = threadIdx.x;
  const int lane = tid & 31;
  const int wave = tid >> 5;
  const int nl   = lane & 15;   // N-column (B/C/D) or M-row (A) lane role
  const int h    = lane >> 4;   // lane-half selector
  float* ystage = smem + PW_OFF + wave * PW_SIZE + YST;
  float* hstage = smem + PW_OFF + wave * PW_SIZE + HST;
  float* dtbuf  = smem + PW_OFF + wave * PW_SIZE + DTB;
  const int m0 = (blockIdx.x * WAVES + wave) * 16;  // this wave's row tile base

  // ---- cooperative weight load into LDS, row-pair interleaved so one
  // ds_load_b64 yields one B-fragment {W[2p][n], W[2p+1][n]} ----
  __builtin_prefetch(W1, 0, 1);
  __builtin_prefetch(W2, 0, 1);
  for (int i = tid; i < 32 * 512; i += BLOCK) {   // W1: 64x512
    int p = i >> 9, n = i & 511;
    v2f t;
    t.x = W1[(2 * p) * 512 + n];
    t.y = W1[(2 * p + 1) * 512 + n];
    *(v2f*)&w1i[2 * i] = t;
  }
  for (int i = tid; i < 256 * 64; i += BLOCK) {   // W2: 512x64
    int p = i >> 6, n = i & 63;
    v2f t;
    t.x = W2[(2 * p) * 64 + n];
    t.y = W2[(2 * p + 1) * 64 + n];
    *(v2f*)&w2i[2 * i] = t;
  }
  for (int i = tid; i < 512; i += BLOCK) b1s[i] = b1[i];
  for (int i = tid; i < 64; i += BLOCK)  b2s[i] = b2[i];

  // ---- per-wave dt table: dt[j][rr] = ts[row][j+1] - ts[row][j] ----
  const int S = T - 1;
  for (int i = lane; i < S * 16; i += 32) {
    int j = i >> 4, rr = i & 15;
    int row = m0 + rr;
    dtbuf[i] = ts[row * T + j + 1] - ts[row * T + j];
  }
  __syncthreads();

  // per-lane base pointers for B-fragment streams
  const float* w1lane = w1i + 2 * (h * 512 + nl);  // + k4*2048 + col16*32
  const float* w2lane = w2i + 2 * (h * 64 + nl);   // + p*128 + c*32

  // ---- state in registers, C-layout: y[c][r] = y[row=m0+r+8h][col=16c+nl] ----
  v8f y[4], f1[4], f2[4], f3[4];
#pragma unroll
  for (int r = 0; r < 8; r++) {
    int row = m0 + r + 8 * h;
    float mk = masks[row];
#pragma unroll
    for (int c = 0; c < 4; c++) y[c][r] = hxs[row * 64 + 16 * c + nl] * mk;
  }
#pragma unroll
  for (int c = 0; c < 4; c++) { f1[c] = splat8(0.0f); f2[c] = splat8(0.0f); f3[c] = splat8(0.0f); }

  // ---- 127 sequential AB steps, all in-kernel ----
  for (int j = 0; j < S; j++) {
    // stage y row-major so it can be re-read in A-layout
#pragma unroll
    for (int c = 0; c < 4; c++)
#pragma unroll
      for (int r = 0; r < 8; r++)
        ystage[(r + 8 * h) * 64 + 16 * c + nl] = y[c][r];

    // A-fragments for GEMM1 (K = 0..63): lane m=nl, K pair at 4k4 + 2h
    v2f af[16];
#pragma unroll
    for (int k4 = 0; k4 < 16; k4++)
      af[k4] = *(const v2f*)&ystage[nl * 64 + 4 * k4 + 2 * h];

    // f accumulator starts at bias b2 (4 independent WMMA chains)
    v8f facc[4];
#pragma unroll
    for (int c = 0; c < 4; c++) facc[c] = splat8(b2s[16 * c + nl]);

    for (int j1 = 0; j1 < 32; j1 += 2) {   // two hidden n-tiles per iteration
      v8f hA = splat8(b1s[16 * j1 + nl]);
      v8f hB = splat8(b1s[16 * j1 + 16 + nl]);

      const float* bbase = w1lane + 32 * j1;   // col stride: 16 cols = 32 floats
      // software-pipelined B-fragment stream, 1 k-step ahead; the A/B column
      // pair sits 128B apart -> ds_load_2addr_b64 mergeable
      v2f bA = *(const v2f*)&bbase[0];
      v2f bB = *(const v2f*)&bbase[32];
      v2f pA = bA, pB = bB;
#pragma unroll
      for (int k4 = 0; k4 < 16; k4++) {
        if (k4 + 1 < 16) {
          pA = *(const v2f*)&bbase[2048 * (k4 + 1)];
          pB = *(const v2f*)&bbase[2048 * (k4 + 1) + 32];
        }
        hA = wmma4(af[k4], bA, hA);      // two independent accumulator chains,
        hB = wmma4(af[k4], bB, hB);      // shared A-fragment
        bA = pA;
        bB = pB;
      }
#pragma unroll
      for (int r = 0; r < 8; r++) { hA[r] = fast_tanh(hA[r]); hB[r] = fast_tanh(hB[r]); }

      // C-layout -> A-layout bounce via per-wave LDS (in-order within wave)
#pragma unroll
      for (int r = 0; r < 8; r++) {
        hstage[(r + 8 * h) * 16 + nl]       = hA[r];
        hstage[256 + (r + 8 * h) * 16 + nl] = hB[r];
      }
      v2f a2[8];
#pragma unroll
      for (int kk = 0; kk < 4; kk++) {
        a2[kk]     = *(const v2f*)&hstage[nl * 16 + 4 * kk + 2 * h];
        a2[4 + kk] = *(const v2f*)&hstage[256 + nl * 16 + 4 * kk + 2 * h];
      }

      // GEMM2: tiles contribute K=[16*j1, 16*j1+32) in 8 chunks of 4.
      // B-fragment pair index p = 8*j1 + 8*(t>>2) + 2*(t&3) + h, prefetch 1 t ahead.
      const float* gbase = w2lane + 1024 * j1;   // p*128 with p offset folded
      v2f g0 = *(const v2f*)&gbase[0];
      v2f g1 = *(const v2f*)&gbase[32];
      v2f g2 = *(const v2f*)&gbase[64];
      v2f g3 = *(const v2f*)&gbase[96];
#pragma unroll
      for (int t = 0; t < 8; t++) {
        v2f n0 = g0, n1 = g1, n2 = g2, n3 = g3;
        if (t + 1 < 8) {
          int off = 1024 * ((t + 1) >> 2) + 256 * ((t + 1) & 3);
          n0 = *(const v2f*)&gbase[off];
          n1 = *(const v2f*)&gbase[off + 32];
          n2 = *(const v2f*)&gbase[off + 64];
          n3 = *(const v2f*)&gbase[off + 96];
        }
        facc[0] = wmma4(a2[t], g0, facc[0]);
        facc[1] = wmma4(a2[t], g1, facc[1]);
        facc[2] = wmma4(a2[t], g2, facc[2]);
        facc[3] = wmma4(a2[t], g3, facc[3]);
        g0 = n0; g1 = n1; g2 = n2; g3 = n3;
      }
    }

    // dt for my 8 rows (16B-aligned LDS loads)
    float4 d0 = *(const float4*)&dtbuf[j * 16 + 8 * h];
    float4 d1 = *(const float4*)&dtbuf[j * 16 + 8 * h + 4];
    float dtv[8] = {d0.x, d0.y, d0.z, d0.w, d1.x, d1.y, d1.z, d1.w};

    if (j >= 3) {
#pragma unroll
      for (int c = 0; c < 4; c++)
#pragma unroll
        for (int r = 0; r < 8; r++)
          y[c][r] += dtv[r] * ((55.f * facc[c][r] - 59.f * f1[c][r] +
                                37.f * f2[c][r] - 9.f * f3[c][r]) * (1.0f / 24.0f));
    } else if (j == 2) {
#pragma unroll
      for (int c = 0; c < 4; c++)
#pragma unroll
        for (int r = 0; r < 8; r++)
          y[c][r] += dtv[r] * ((23.f * facc[c][r] - 16.f * f1[c][r] +
                                5.f * f2[c][r]) * (1.0f / 12.0f));
    } else if (j == 1) {
#pragma unroll
      for (int c = 0; c < 4; c++)
#pragma unroll
        for (int r = 0; r < 8; r++)
          y[c][r] += dtv[r] * ((3.f * facc[c][r] - f1[c][r]) * 0.5f);
    } else {
#pragma unroll
      for (int c = 0; c < 4; c++)
#pragma unroll
        for (int r = 0; r < 8; r++)
          y[c][r] += dtv[r] * facc[c][r];
    }
#pragma unroll
    for (int c = 0; c < 4; c++) { f3[c] = f2[c]; f2[c] = f1[c]; f1[c] = facc[c]; }
  }

  // ---- t0 == 0 passthrough + store ----
#pragma unroll
  for (int r = 0; r < 8; r++) {
    int row = m0 + r + 8 * h;
    bool pass = (ts[row * T] == 0.0f);
    float mk = pass ? masks[row] : 0.0f;
#pragma unroll
    for (int c = 0; c < 4; c++) {
      float v = y[c][r];
      if (pass) v = hxs[row * 64 + 16 * c + nl] * mk;
      out[row * 64 + 16 * c + nl] = v;
    }
  }
}

extern "C" void kernel_launch(void* const* d_in, const int* in_sizes, int n_in,
                              void* d_out, int out_size, void* d_ws, size_t ws_size,
                              hipStream_t stream) {
  const float* ts    = (const float*)d_in[0];   // [N, T]
  const float* hxs   = (const float*)d_in[1];   // [N, 64]
  const float* masks = (const float*)d_in[2];   // [N, 1]
  const float* W1    = (const float*)d_in[3];   // [64, 512]
  const float* b1    = (const float*)d_in[4];   // [512]
  const float* W2    = (const float*)d_in[5];   // [512, 64]
  const float* b2    = (const float*)d_in[6];   // [64]
  float* out = (float*)d_out;                   // [N, 64]

  int N = in_sizes[1] / 64;
  int T = in_sizes[0] / N;

  size_t smem_bytes = (size_t)SMEM_FLOATS * sizeof(float);  // 321,792 B
  (void)hipFuncSetAttribute((const void*)node_ab4_kernel,
                            hipFuncAttributeMaxDynamicSharedMemorySize,
                            (int)smem_bytes);

  int rows_per_block = WAVES * 16;
  int grid = (N + rows_per_block - 1) / rows_per_block;   // 128 blocks for N=8192
  node_ab4_kernel<<<grid, BLOCK, smem_bytes, stream>>>(ts, hxs, masks, W1, b1,
                                                       W2, b2, out, N, T);
}